// NetGIN_54683523612723
// MI455X (gfx1250) — compile-verified
//
#include <hip/hip_runtime.h>
#include <hip/hip_bf16.h>
#include <math.h>

#define N_NODES   100000
#define N_EDGES   1600000
#define N_GRAPHS  1000
#define NFEAT     128
#define DIM       64

typedef float v2f __attribute__((ext_vector_type(2)));
typedef float v8f __attribute__((ext_vector_type(8)));

// ---------------------------------------------------------------------------
// small helpers
// ---------------------------------------------------------------------------
__device__ __forceinline__ void atomic_add_f32(float* p, float v) {
  // lowers to global_atomic_add_f32 on gfx1250 (L2-resident agg/score buffers)
  unsafeAtomicAdd(p, v);
}

__global__ void zero_f_kernel(float* __restrict__ p, int n) {
  int i = blockIdx.x * blockDim.x + threadIdx.x;
  if (i < n) p[i] = 0.0f;
}

__global__ void copy_f4_kernel(const float4* __restrict__ in,
                               float4* __restrict__ out, int n4) {
  int i = blockIdx.x * blockDim.x + threadIdx.x;
  if (i < n4) out[i] = in[i];
}

__global__ void counts_kernel(const long long* __restrict__ batch,
                              float* __restrict__ counts, int n) {
  int i = blockIdx.x * blockDim.x + threadIdx.x;
  if (i < n) atomic_add_f32(&counts[(int)batch[i]], 1.0f);
}

// ---------------------------------------------------------------------------
// edge-parallel scatter-add:  agg[dst] += h[src]   (one thread per edge-float)
// ---------------------------------------------------------------------------
template <int D>
__global__ void scatter_kernel(const long long* __restrict__ src,
                               const long long* __restrict__ dst,
                               const float* __restrict__ h,
                               float* __restrict__ agg) {
  long long tid = (long long)blockIdx.x * blockDim.x + threadIdx.x;
  const long long total = (long long)N_EDGES * D;
  if (tid >= total) return;
  int e = (int)(tid / D);          // D is a power of two -> shift
  int f = (int)(tid & (D - 1));
  int s = (int)src[e];
  int d = (int)dst[e];
  atomic_add_f32(agg + (long long)d * D + f, h[(long long)s * D + f]);
}

// ---------------------------------------------------------------------------
// fused 2-layer MLP via fp32 WMMA (V_WMMA_F32_16X16X4_F32).
// One wave per 16-node tile. 4 waves / block.
//   hout = relu( relu(agg @ W1 + b1) @ W2 + b2 )     [16 x 64 per tile]
// ---------------------------------------------------------------------------
template <int KDIM>
__global__ void __launch_bounds__(128)
gin_mlp_kernel(const float* __restrict__ agg,   // [N, KDIM]
               const float* __restrict__ W1,    // [KDIM, 64]
               const float* __restrict__ b1,    // [64]
               const float* __restrict__ W2,    // [64, 64]
               const float* __restrict__ b2,    // [64]
               float* __restrict__ hout,        // [N, 64]
               int nTiles) {
  __shared__ float lds[4 * 16 * (KDIM + 64)];
  const int lane = threadIdx.x & 31;
  const int wave = threadIdx.x >> 5;
  const int tile = blockIdx.x * 4 + wave;
  const int col  = lane & 15;          // column / row-in-tile for frag layouts
  const bool hi  = lane >= 16;
  const int ksel = hi ? 2 : 0;         // A/B fragment K selector (ISA 7.12.2)
  const bool active = tile < nTiles;   // wave-uniform -> EXEC all-ones in WMMA

  float* Ash = lds + wave * 16 * (KDIM + 64);  // staged activation tile
  float* Ish = Ash + 16 * KDIM;                // relu intermediate [16][64]

  if (active) {
    const float* srcp = agg + (long long)tile * 16 * KDIM;
    for (int i = lane; i < 16 * KDIM; i += 32) Ash[i] = srcp[i];
  }
  __syncthreads();

  v8f acc[4];

  if (active) {
    // ---- GEMM1: [16xKDIM] @ [KDIMx64] + b1 ----
#pragma unroll
    for (int nt = 0; nt < 4; ++nt) {
      float bv = b1[nt * 16 + col];    // C layout: lane -> column, VGPR -> row
#pragma unroll
      for (int v = 0; v < 8; ++v) acc[nt][v] = bv;
    }
#pragma unroll 4
    for (int k = 0; k < KDIM; k += 4) {
      v2f a;
      a.x = Ash[col * KDIM + k + ksel];
      a.y = Ash[col * KDIM + k + ksel + 1];
#pragma unroll
      for (int nt = 0; nt < 4; ++nt) {
        v2f b;
        b.x = W1[(k + ksel) * 64 + nt * 16 + col];
        b.y = W1[(k + ksel + 1) * 64 + nt * 16 + col];
        acc[nt] = __builtin_amdgcn_wmma_f32_16x16x4_f32(
            false, a, false, b, (short)0, acc[nt], false, false);
      }
    }
    // relu -> LDS intermediate [16][64]
#pragma unroll
    for (int nt = 0; nt < 4; ++nt) {
#pragma unroll
      for (int v = 0; v < 8; ++v) {
        int row = v + (hi ? 8 : 0);
        float val = acc[nt][v];
        Ish[row * 64 + nt * 16 + col] = val > 0.0f ? val : 0.0f;
      }
    }
  }
  __syncthreads();

  if (active) {
    // ---- GEMM2: [16x64] @ [64x64] + b2 ----
#pragma unroll
    for (int nt = 0; nt < 4; ++nt) {
      float bv = b2[nt * 16 + col];
#pragma unroll
      for (int v = 0; v < 8; ++v) acc[nt][v] = bv;
    }
#pragma unroll 4
    for (int k = 0; k < 64; k += 4) {
      v2f a;
      a.x = Ish[col * 64 + k + ksel];
      a.y = Ish[col * 64 + k + ksel + 1];
#pragma unroll
      for (int nt = 0; nt < 4; ++nt) {
        v2f b;
        b.x = W2[(k + ksel) * 64 + nt * 16 + col];
        b.y = W2[(k + ksel + 1) * 64 + nt * 16 + col];
        acc[nt] = __builtin_amdgcn_wmma_f32_16x16x4_f32(
            false, a, false, b, (short)0, acc[nt], false, false);
      }
    }
    // relu -> global
    float* dst = hout + (long long)tile * 16 * 64;
#pragma unroll
    for (int nt = 0; nt < 4; ++nt) {
#pragma unroll
      for (int v = 0; v < 8; ++v) {
        int row = v + (hi ? 8 : 0);
        float val = acc[nt][v];
        dst[row * 64 + nt * 16 + col] = val > 0.0f ? val : 0.0f;
      }
    }
  }
}

// ---------------------------------------------------------------------------
// score[g] += dot(h[i], L) / max(counts[g],1)   (one thread per node)
// ---------------------------------------------------------------------------
__global__ void score_kernel(const float* __restrict__ h,
                             const long long* __restrict__ batch,
                             const float* __restrict__ counts,
                             const float* __restrict__ Lrow,
                             float* __restrict__ score, int n) {
  int i = blockIdx.x * blockDim.x + threadIdx.x;
  if (i >= n) return;
  int g = (int)batch[i];
  const float* hr = h + (long long)i * DIM;
  float dot = 0.0f;
#pragma unroll
  for (int f = 0; f < DIM; ++f) dot += hr[f] * Lrow[f];
  float c = counts[g];
  if (c < 1.0f) c = 1.0f;
  atomic_add_f32(&score[g], dot / c);
}

__global__ void sigmoid_kernel(const float* __restrict__ score,
                               float* __restrict__ out, int n) {
  int i = blockIdx.x * blockDim.x + threadIdx.x;
  if (i < n) out[i] = 1.0f / (1.0f + expf(-score[i]));
}

// ---------------------------------------------------------------------------
extern "C" void kernel_launch(void* const* d_in, const int* in_sizes, int n_in,
                              void* d_out, int out_size, void* d_ws, size_t ws_size,
                              hipStream_t stream) {
  const float*      x     = (const float*)d_in[0];
  const long long*  eidx  = (const long long*)d_in[1];
  const long long*  batch = (const long long*)d_in[2];
  const float*      W1a   = (const float*)d_in[3];
  const float*      b1a   = (const float*)d_in[4];
  const float*      W1b   = (const float*)d_in[5];
  const float*      b1b   = (const float*)d_in[6];
  const float*      Ws1   = (const float*)d_in[7];   // [4,64,64]
  const float*      bs1   = (const float*)d_in[8];   // [4,64]
  const float*      Ws2   = (const float*)d_in[9];   // [4,64,64]
  const float*      bs2   = (const float*)d_in[10];  // [4,64]
  const float*      L     = (const float*)d_in[11];  // [5,64]
  float*            out   = (float*)d_out;

  const long long* esrc = eidx;
  const long long* edst = eidx + N_EDGES;

  // workspace layout
  float* agg    = (float*)d_ws;                       // [N, 128] (reused as [N,64])
  float* h      = agg + (size_t)N_NODES * NFEAT;      // [N, 64]
  float* counts = h + (size_t)N_NODES * DIM;          // [G]
  float* score  = counts + N_GRAPHS;                  // [G]

  const int nTiles = N_NODES / 16;                    // 6250
  const int mlpBlocks = (nTiles + 3) / 4;

  // counts + score init
  zero_f_kernel<<<(2 * N_GRAPHS + 255) / 256, 256, 0, stream>>>(counts, 2 * N_GRAPHS);
  counts_kernel<<<(N_NODES + 255) / 256, 256, 0, stream>>>(batch, counts, N_NODES);

  // ---- layer 0 (feature dim 128) ----
  {
    int n4 = N_NODES * NFEAT / 4;
    copy_f4_kernel<<<(n4 + 255) / 256, 256, 0, stream>>>((const float4*)x, (float4*)agg, n4);
    long long total = (long long)N_EDGES * NFEAT;
    scatter_kernel<NFEAT><<<(int)((total + 255) / 256), 256, 0, stream>>>(esrc, edst, x, agg);
    gin_mlp_kernel<NFEAT><<<mlpBlocks, 128, 0, stream>>>(agg, W1a, b1a, W1b, b1b, h, nTiles);
    score_kernel<<<(N_NODES + 255) / 256, 256, 0, stream>>>(h, batch, counts, L, score, N_NODES);
  }

  // ---- layers 1..4 (feature dim 64) ----
  for (int k = 0; k < 4; ++k) {
    int n4 = N_NODES * DIM / 4;
    copy_f4_kernel<<<(n4 + 255) / 256, 256, 0, stream>>>((const float4*)h, (float4*)agg, n4);
    long long total = (long long)N_EDGES * DIM;
    scatter_kernel<DIM><<<(int)((total + 255) / 256), 256, 0, stream>>>(esrc, edst, h, agg);
    gin_mlp_kernel<DIM><<<mlpBlocks, 128, 0, stream>>>(
        agg, Ws1 + (size_t)k * DIM * DIM, bs1 + (size_t)k * DIM,
        Ws2 + (size_t)k * DIM * DIM, bs2 + (size_t)k * DIM, h, nTiles);
    score_kernel<<<(N_NODES + 255) / 256, 256, 0, stream>>>(h, batch, counts,
                                                            L + (size_t)(k + 1) * DIM,
                                                            score, N_NODES);
  }

  sigmoid_kernel<<<(N_GRAPHS + 255) / 256, 256, 0, stream>>>(score, out, N_GRAPHS);
}